// PrtNN_1460288881510
// MI455X (gfx1250) — compile-verified
//
#include <hip/hip_runtime.h>
#include <hip/hip_bf16.h>

typedef __attribute__((ext_vector_type(2))) float v2f;
typedef __attribute__((ext_vector_type(8))) float v8f;

#define BATCH 32
#define NHITS 200
#define DIM1  350
#define OUTN  5

// One wave32 per batch row. Dedup the 200 scatter hits, gather the hit rows of
// W into LDS (16-wide, zero padded), then reduce them with exact f32 WMMA:
//   D = ones(16x4) x B(4x16)  =>  every row of D = column sums of B.
// Accumulated over 50 chunks of 4 hits; lanes 0..4 of C vgpr0 hold out[b,0..4].
__global__ __launch_bounds__(32) void prtnn_scatter_dense_kernel(
    const int* __restrict__ x,      // [32, 200, 3] (b, i0, i1)
    const float* __restrict__ W,    // [2150400, 5]
    const float* __restrict__ bias, // [5]
    float* __restrict__ out)        // [32, 5]
{
  __shared__ int   keys[NHITS];
  __shared__ float gv[NHITS * 16];  // gathered W rows, cols 5..15 zeroed

  const int b = blockIdx.x;
  const int t = threadIdx.x;

  // Phase A: compute flat feature keys for this row's 200 hits.
  for (int j = t; j < NHITS; j += 32) {
    const int base = (b * NHITS + j) * 3;
    const int i0 = x[base + 1];
    const int i1 = x[base + 2];
    keys[j] = i0 * DIM1 + i1;
  }
  __syncthreads();

  // Phase B: dedup (scatter .set(1.0) counts each position once) and gather
  // the 5-float W row for each first-occurrence hit; zeros for dups/padding.
  for (int j = t; j < NHITS; j += 32) {
    const int key = keys[j];
    bool dup = false;
    for (int jj = 0; jj < j; ++jj) dup |= (keys[jj] == key);
    const float* wrow = W + key * OUTN;
#pragma unroll
    for (int o = 0; o < 16; ++o) {
      gv[j * 16 + o] = (!dup && o < OUTN) ? wrow[o] : 0.0f;
    }
  }
  __syncthreads();

  // Phase C: exact f32 WMMA reduction. A = all-ones (layout-independent), so
  // D[m][n] = sum_k B[k][n]; only lane%16 == N matters for B placement.
  const int n  = t & 15;          // column (N)
  const int k0 = (t >> 4) * 2;    // which pair of K rows this lane half holds
  v2f a; a.x = 1.0f; a.y = 1.0f;
  v8f c = {};
  for (int chunk = 0; chunk < NHITS / 4; ++chunk) {
    const int j0 = chunk * 4;
    v2f bm;
    bm.x = gv[(j0 + k0) * 16 + n];      // 32 lanes -> 32 distinct LDS banks
    bm.y = gv[(j0 + k0 + 1) * 16 + n];
    c = __builtin_amdgcn_wmma_f32_16x16x4_f32(false, a, false, bm,
                                              (short)0, c, false, false);
  }

  // Phase D: C/D vgpr0, lane L<16 = (M=0, N=L); all rows identical.
  if (t < OUTN) {
    out[b * OUTN + t] = c[0] + bias[t];
  }
}

extern "C" void kernel_launch(void* const* d_in, const int* in_sizes, int n_in,
                              void* d_out, int out_size, void* d_ws, size_t ws_size,
                              hipStream_t stream) {
  (void)in_sizes; (void)n_in; (void)out_size; (void)d_ws; (void)ws_size;
  const int*   x    = (const int*)d_in[0];    // [32,200,3] int32
  const float* W    = (const float*)d_in[1];  // [2150400,5] f32
  const float* bias = (const float*)d_in[2];  // [5] f32
  float*       out  = (float*)d_out;          // [32,5] f32

  prtnn_scatter_dense_kernel<<<BATCH, 32, 0, stream>>>(x, W, bias, out);
}